// AngularBasis_12532714570335
// MI455X (gfx1250) — compile-verified
//
#include <hip/hip_runtime.h>
#include <math.h>

typedef __attribute__((ext_vector_type(2))) float v2f;
typedef __attribute__((ext_vector_type(8))) float v8f;

namespace {

constexpr double sqrt_ce(double v) {
  double g = v > 1.0 ? v : 1.0;
  for (int i = 0; i < 80; ++i) g = 0.5 * (g + v / g);
  return g;
}
constexpr double fact_ce(int n) {
  double r = 1.0;
  for (int i = 2; i <= n; ++i) r *= (double)i;
  return r;
}

// CP[kk][col] = { C[2kk][col], C[2kk+1][col] } where C[k][col] is the
// coefficient of x^k for output column col (col = l*l + l + m) with the full
// normalization folded in.  K padded 9->12 (6 pairs), cols padded 81->96.
// tIdx[col]: modulation index: 0 -> 1.0, am -> s^am*cos(am*phi),
//            8+am -> s^am*sin(am*phi).
struct alignas(16) CTab {
  float CP[6][96][2];
  int   tIdx[96];
};

constexpr CTab make_ctab() {
  CTab t{};
  // q[l][m][k]: polynomial part of P_l^m after factoring out s^m
  double q[9][9][9] = {};
  q[0][0][0] = 1.0;
  for (int m = 1; m <= 8; ++m)
    q[m][m][0] = -(2.0 * m - 1.0) * q[m - 1][m - 1][0];
  for (int m = 0; m <= 7; ++m)
    for (int k = 1; k <= 8; ++k)
      q[m + 1][m][k] = (2.0 * m + 1.0) * q[m][m][k - 1];
  for (int m = 0; m <= 6; ++m)
    for (int l = m + 2; l <= 8; ++l)
      for (int k = 0; k <= 8; ++k) {
        double t1 = (k > 0) ? (2.0 * l - 1.0) * q[l - 1][m][k - 1] : 0.0;
        double t2 = (double)(l + m - 1) * q[l - 2][m][k];
        q[l][m][k] = (t1 - t2) / (double)(l - m);
      }
  const double pi = 3.14159265358979323846;
  const double s2 = sqrt_ce(2.0);
  for (int l = 0; l <= 8; ++l)
    for (int m = -l; m <= l; ++m) {
      int am = (m < 0) ? -m : m;
      int col = l * l + l + m;
      double sgn = (am & 1) ? -1.0 : 1.0;  // (-1)^m == (-1)^am for +/-1 base
      double coef = sgn * s2 *
          sqrt_ce((2.0 * l + 1.0) / (4.0 * pi) * fact_ce(l - am) / fact_ce(l + am));
      if (m == 0) coef = coef / s2;        // ang = 1/sqrt(2) for m==0
      for (int k = 0; k <= 8; ++k) {
        double c = coef * q[l][am][k];
        t.CP[k >> 1][col][k & 1] = (float)c;
      }
      t.tIdx[col] = (m == 0) ? 0 : ((m > 0) ? am : 8 + am);
    }
  return t;
}

}  // namespace

__constant__ CTab g_ct = make_ctab();

// Epilogue + WMMA tile loop.  GUARD=false: dense stores (full 16-point tile);
// GUARD=true: per-row bound checks (only tail waves of the last block take this).
// In both paths EXEC is all-ones at every WMMA (branch is wave-uniform).
template <bool GUARD>
__device__ __forceinline__ void sh_tiles(
    const float* s_cp, const int* s_ti, const float* modp,
    float* __restrict__ outp, float x, float pw0, float pw1, float pw2,
    int lp, int half, int rowsLeft) {
  // Per-lane B base: v2f index = half*96 + lp; per-(t,tile) offsets are
  // compile-time immediates: (t*192 + ctile*16) v2f = t*1536 + ctile*128 B.
  const v2f* cpp = (const v2f*)s_cp + (half * 96 + lp);

#pragma unroll
  for (int ctile = 0; ctile < 6; ++ctile) {
    v2f b0 = cpp[0 * 192 + ctile * 16];
    v2f b1 = cpp[1 * 192 + ctile * 16];
    v2f b2 = cpp[2 * 192 + ctile * 16];

    v8f acc = {};
    {
      v2f a = { pw0, pw0 * x };
      acc = __builtin_amdgcn_wmma_f32_16x16x4_f32(false, a, false, b0,
                                                  (short)0, acc, false, false);
    }
    {
      v2f a = { pw1, pw1 * x };
      acc = __builtin_amdgcn_wmma_f32_16x16x4_f32(false, a, false, b1,
                                                  (short)0, acc, false, false);
    }
    {
      v2f a = { pw2, pw2 * x };
      acc = __builtin_amdgcn_wmma_f32_16x16x4_f32(false, a, false, b2,
                                                  (short)0, acc, false, false);
    }

    const int t = s_ti[ctile * 16 + lp];
    // ctile is an unroll-time constant: folds to `true` for ctile<5 and to
    // `lp == 0` (col 80) for ctile==5.
    const bool colValid = (ctile < 5) || (lp == 0);
    if (colValid) {
#pragma unroll
      for (int j = 0; j < 8; ++j) {
        if (!GUARD || j < rowsLeft) {
          float mod = modp[j * 17 + t];
          __builtin_nontemporal_store(acc[j] * mod, outp + j * 81 + ctile * 16);
        }
      }
    }
  }
}

// One wave processes a 16-point tile. 256 threads = 8 waves = 128 points/block.
__global__ __launch_bounds__(256) void sh_wmma_kernel(
    const float* __restrict__ gx, const float* __restrict__ gphi,
    float* __restrict__ out, int n) {
  __shared__ float4 s_cp4[6 * 96 * 2 / 4];  // packed B pairs (4608 B)
  __shared__ int    s_ti[96];               // modulation index per column
  __shared__ float  s_mod[8][16][17];       // per-wave, per-point modulation

  const int tid  = threadIdx.x;
  const int lane = tid & 31;
  const int wave = tid >> 5;
  const int half = lane >> 4;               // K half for A/B layout
  const int lp   = lane & 15;               // A: row (point); B/D: column
  const int base = (blockIdx.x * 8 + wave) * 16;

  // Stage constants into LDS with b128 transfers.
  {
    const float4* src4 = (const float4*)&g_ct.CP[0][0][0];
    for (int i = tid; i < 6 * 96 * 2 / 4; i += 256) s_cp4[i] = src4[i];
    if (tid < 96) s_ti[tid] = g_ct.tIdx[tid];
  }

  int p  = base + lp;
  int pc = (p < n) ? p : (n - 1);           // clamp tail loads
  float x  = gx[pc];
  float ph = gphi[pc];

  // Per-point modulation factors (one writer lane per point).
  if (half == 0) {
    float s = sqrtf((1.0f - x) * (1.0f + x));
    float sn, cs;
    sincosf(ph, &sn, &cs);
    float sm = 0.0f, cm = 1.0f, sp = 1.0f;
    s_mod[wave][lp][0] = 1.0f;
#pragma unroll
    for (int k = 1; k <= 8; ++k) {
      float ns = sm * cs + cm * sn;         // sin(k*phi)
      float nc = cm * cs - sm * sn;         // cos(k*phi)
      sm = ns; cm = nc; sp *= s;
      s_mod[wave][lp][k]     = sp * cm;
      s_mod[wave][lp][8 + k] = sp * sm;
    }
  }
  __syncthreads();

  // Powers of x for A fragments: a_t = { x^(4t+2h), x^(4t+2h+1) }.
  const float x2 = x * x;
  const float x4 = x2 * x2;
  const float x8 = x4 * x4;
  const float u   = half ? x2 : 1.0f;       // x^(2h)
  const float pw0 = u;
  const float pw1 = u * x4;
  const float pw2 = u * x8;

  float* outp = out + (size_t)(base + 8 * half) * 81 + lp;
  const float* modp = &s_mod[wave][8 * half][0];
  const float* s_cp = (const float*)s_cp4;

  if (base + 16 <= n) {                     // wave-uniform fast path
    sh_tiles<false>(s_cp, s_ti, modp, outp, x, pw0, pw1, pw2, lp, half, 8);
  } else {                                  // tail wave(s) only
    int rowsLeft = n - base - 8 * half;     // may be <= 0
    sh_tiles<true>(s_cp, s_ti, modp, outp, x, pw0, pw1, pw2, lp, half, rowsLeft);
  }
}

extern "C" void kernel_launch(void* const* d_in, const int* in_sizes, int n_in,
                              void* d_out, int out_size, void* d_ws, size_t ws_size,
                              hipStream_t stream) {
  const float* gx   = (const float*)d_in[0];
  const float* gphi = (const float*)d_in[1];
  float* out = (float*)d_out;
  const int n = in_sizes[0];
  const int blocks = (n + 127) / 128;       // 128 points per block (8 waves x 16)
  sh_wmma_kernel<<<blocks, 256, 0, stream>>>(gx, gphi, out, n);
  (void)n_in; (void)out_size; (void)d_ws; (void)ws_size;
}